// GATBasedMolecularGraphResNet_54872502173932
// MI455X (gfx1250) — compile-verified
//
#include <hip/hip_runtime.h>
#include <hip/hip_bf16.h>
#include <math.h>

// ---------------------------------------------------------------------------
// GIN GNN for MI455X (gfx1250, wave32). Dense GEMMs use V_WMMA_F32_16X16X4_F32
// (fp32 matrix pipe -> no precision loss vs the fp32 reference). Irregular
// aggregation / pooling use fp32 global atomics (bandwidth-bound anyway).
// ---------------------------------------------------------------------------

#define N_NODES   100000
#define N_EDGES   1600000
#define N_GRAPHS  4096
#define FDIM      128
#define FIN       48
#define GN_EPS    1e-5f
#define SLOPE     0.01f

typedef __attribute__((ext_vector_type(2))) float v2f;
typedef __attribute__((ext_vector_type(8))) float v8f;

// ---------------------------------------------------------------------------
// WMMA GEMM:  C[M x 128] = A[M x K] @ W[K x 128] + bias
// 256 threads = 8 waves. W staged in LDS (<=64KB). Each wave computes one
// 16-row strip x 128 cols using 8 accumulators of v_wmma_f32_16x16x4_f32.
// M is a multiple of 16 (100000 = 6250*16); tile guard is wave-uniform so
// EXEC stays all-ones inside WMMA (ISA requirement).
// ---------------------------------------------------------------------------
__global__ void __launch_bounds__(256)
gemm_wmma_k(const float* __restrict__ A, const float* __restrict__ W,
            const float* __restrict__ bias, float* __restrict__ C,
            int mtiles, int K)
{
    __shared__ float sW[FDIM * FDIM];   // 64 KB max (K<=128)

    const int tid = threadIdx.x;
    const int total = K * FDIM;
    for (int i = tid * 4; i < total; i += 256 * 4) {
        *(float4*)&sW[i] = *(const float4*)&W[i];
    }
    __syncthreads();

    const int wave = tid >> 5;
    const int lane = tid & 31;
    const int tile = blockIdx.x * 8 + wave;
    if (tile >= mtiles) return;               // wave-uniform

    const int row0 = tile * 16;
    const int m    = lane & 15;               // A row within tile / B col within tile
    const int kk   = (lane >> 4) << 1;        // 0 or 2 (K sub-offset per ISA layout)

    v8f acc[8];
    #pragma unroll
    for (int t = 0; t < 8; ++t) acc[t] = (v8f){};

    const float* Arow = A + (size_t)(row0 + m) * K;

    for (int k0 = 0; k0 < K; k0 += 4) {
        v2f a;
        a.x = Arow[k0 + kk];
        a.y = Arow[k0 + kk + 1];
        #pragma unroll
        for (int t = 0; t < 8; ++t) {
            const int n0 = t * 16;
            v2f b;
            b.x = sW[(k0 + kk)     * FDIM + n0 + m];
            b.y = sW[(k0 + kk + 1) * FDIM + n0 + m];
            acc[t] = __builtin_amdgcn_wmma_f32_16x16x4_f32(
                false, a, false, b, (short)0, acc[t], false, false);
        }
    }

    // C/D layout: VGPR r -> lanes 0-15: M=r, lanes 16-31: M=r+8; N = lane&15
    const int mb = (lane >> 4) * 8;
    #pragma unroll
    for (int t = 0; t < 8; ++t) {
        const int n0 = t * 16;
        const float bv = bias[n0 + m];
        #pragma unroll
        for (int r = 0; r < 8; ++r) {
            C[(size_t)(row0 + mb + r) * FDIM + n0 + m] = acc[t][r] + bv;
        }
    }
}

// ---------------------------------------------------------------------------
// Edge aggregation: out[dst] += feat[src]   (out pre-initialized to h -> (1+eps)x + sum)
// One thread per (edge, 4-feature group): float4 gather + 4 fp32 atomics.
// ---------------------------------------------------------------------------
__global__ void scatter_add_k(const float* __restrict__ feat,
                              const int* __restrict__ src,
                              const int* __restrict__ dst,
                              float* __restrict__ out,
                              int groups /* F/4 */)
{
    const unsigned long long t =
        (unsigned long long)blockIdx.x * blockDim.x + threadIdx.x;
    const unsigned long long total = (unsigned long long)N_EDGES * groups;
    if (t >= total) return;
    const int e  = (int)(t / groups);
    const int fg = (int)(t % groups);
    const int F  = groups * 4;
    const int s  = src[e];
    const int d  = dst[e];
    const float4 v = *(const float4*)(feat + (size_t)s * F + fg * 4);
    float* o = out + (size_t)d * F + fg * 4;
    atomicAdd(o + 0, v.x);
    atomicAdd(o + 1, v.y);
    atomicAdd(o + 2, v.z);
    atomicAdd(o + 3, v.w);
}

// ---------------------------------------------------------------------------
// GraphNorm (single-graph: mean/var over ALL rows, per feature), 3 passes.
// ---------------------------------------------------------------------------
#define COL_ROWS 256
__global__ void colsum_k(const float* __restrict__ x, float* __restrict__ sums,
                         int nrows)
{
    const int f  = threadIdx.x;                 // 128
    const int r0 = blockIdx.x * COL_ROWS;
    const int r1 = min(r0 + COL_ROWS, nrows);
    float s = 0.f;
    for (int r = r0; r < r1; ++r) s += x[(size_t)r * FDIM + f];
    atomicAdd(&sums[f], s);
}

__global__ void colvar_k(const float* __restrict__ x, const float* __restrict__ sums,
                         const float* __restrict__ mscale, float* __restrict__ vsums,
                         int nrows)
{
    const int f  = threadIdx.x;
    const float mean = sums[f] * (1.0f / (float)nrows);
    const float sm   = mscale[f] * mean;
    const int r0 = blockIdx.x * COL_ROWS;
    const int r1 = min(r0 + COL_ROWS, nrows);
    float s = 0.f;
    for (int r = r0; r < r1; ++r) {
        const float t = x[(size_t)r * FDIM + f] - sm;
        s += t * t;
    }
    atomicAdd(&vsums[f], s);
}

#define APPLY_ROWS 64
__global__ void norm_apply_k(float* __restrict__ x,
                             const float* __restrict__ sums,
                             const float* __restrict__ vsums,
                             const float* __restrict__ w,
                             const float* __restrict__ b,
                             const float* __restrict__ mscale,
                             int nrows)
{
    const int f  = threadIdx.x;
    const float inv_n = 1.0f / (float)nrows;
    const float mean  = sums[f] * inv_n;
    const float sm    = mscale[f] * mean;
    const float var   = vsums[f] * inv_n;
    const float inv   = rsqrtf(var + GN_EPS);
    const float wf = w[f], bf = b[f];
    const int r0 = blockIdx.x * APPLY_ROWS;
    const int r1 = min(r0 + APPLY_ROWS, nrows);
    for (int r = r0; r < r1; ++r) {
        float v = (x[(size_t)r * FDIM + f] - sm) * inv * wf + bf;
        v = v > 0.f ? v : SLOPE * v;            // leaky relu fused
        x[(size_t)r * FDIM + f] = v;
    }
}

// ---------------------------------------------------------------------------
// Pooling: per-graph sum / count / max.
// ---------------------------------------------------------------------------
__device__ __forceinline__ void atomicMaxF(float* addr, float val)
{
    if (val >= 0.f) atomicMax((int*)addr, __float_as_int(val));
    else            atomicMin((unsigned int*)addr, __float_as_uint(val));
}

__global__ void fill_neg_inf_k(float* p, int n)
{
    const int i = blockIdx.x * blockDim.x + threadIdx.x;
    if (i < n) p[i] = -INFINITY;
}

__global__ void pool_k(const float* __restrict__ h, const int* __restrict__ batch,
                       float* __restrict__ psum, float* __restrict__ pmax,
                       float* __restrict__ cnt)
{
    const int node = blockIdx.x;
    const int f    = threadIdx.x;               // 128
    const int g    = batch[node];
    const float v  = h[(size_t)node * FDIM + f];
    atomicAdd(&psum[(size_t)g * FDIM + f], v);
    atomicMaxF(&pmax[(size_t)g * FDIM + f], v);
    if (f == 0) atomicAdd(&cnt[g], 1.0f);
}

__global__ void finalize_k(const float* __restrict__ psum,
                           const float* __restrict__ pmax,
                           const float* __restrict__ cnt,
                           float* __restrict__ g)
{
    const int gr = blockIdx.x;                  // 4096
    const int f  = threadIdx.x;                 // 128
    const float c = cnt[gr];
    const float s = psum[(size_t)gr * FDIM + f];
    const float mean = s / fmaxf(c, 1.0f);
    float m = pmax[(size_t)gr * FDIM + f];
    if (m < -3.0e38f) m = 0.f;                  // empty-segment guard (mx was -inf)
    g[(size_t)gr * 384 + f]       = mean;
    g[(size_t)gr * 384 + 128 + f] = s;
    g[(size_t)gr * 384 + 256 + f] = m;
}

// ---------------------------------------------------------------------------
// Readout MLP: [G,384] -> leaky(fc1:384x64) -> fc2:64x3
// ---------------------------------------------------------------------------
__global__ void __launch_bounds__(64)
readout_k(const float* __restrict__ g,
          const float* __restrict__ fc1W, const float* __restrict__ fc1b,
          const float* __restrict__ fc2W, const float* __restrict__ fc2b,
          float* __restrict__ out)
{
    __shared__ float sg[384];
    __shared__ float h1[64];
    const int gr = blockIdx.x;
    const int t  = threadIdx.x;                 // 64
    for (int i = t; i < 384; i += 64) sg[i] = g[(size_t)gr * 384 + i];
    __syncthreads();
    float acc = fc1b[t];
    #pragma unroll 8
    for (int i = 0; i < 384; ++i) acc += sg[i] * fc1W[i * 64 + t];
    h1[t] = acc > 0.f ? acc : SLOPE * acc;
    __syncthreads();
    if (t < 3) {
        float o = fc2b[t];
        #pragma unroll
        for (int i = 0; i < 64; ++i) o += h1[i] * fc2W[i * 3 + t];
        out[(size_t)gr * 3 + t] = o;
    }
}

// ---------------------------------------------------------------------------
// Host-side orchestration
// ---------------------------------------------------------------------------
static inline size_t align256(size_t x) { return (x + 255) & ~(size_t)255; }

extern "C" void kernel_launch(void* const* d_in, const int* in_sizes, int n_in,
                              void* d_out, int out_size, void* d_ws, size_t ws_size,
                              hipStream_t stream)
{
    (void)in_sizes; (void)n_in; (void)out_size; (void)ws_size;

    const float* x       = (const float*)d_in[0];
    /* edge_attr d_in[1] ignored (GIN has no edge features) */
    const float* W1_0    = (const float*)d_in[2];
    const float* b1_0    = (const float*)d_in[3];
    const float* W1_rest = (const float*)d_in[4];   // [3,128,128]
    const float* b1_rest = (const float*)d_in[5];   // [3,128]
    const float* gn_w    = (const float*)d_in[6];   // [4,128]
    const float* gn_b    = (const float*)d_in[7];
    const float* gn_s    = (const float*)d_in[8];
    const float* W2      = (const float*)d_in[9];   // [4,128,128]
    const float* b2      = (const float*)d_in[10];  // [4,128]
    const float* pn_w    = (const float*)d_in[11];  // [3,128]
    const float* pn_b    = (const float*)d_in[12];
    const float* pn_s    = (const float*)d_in[13];
    const float* fc1_W   = (const float*)d_in[14];  // [384,64]
    const float* fc1_b   = (const float*)d_in[15];
    const float* fc2_W   = (const float*)d_in[16];  // [64,3]
    const float* fc2_b   = (const float*)d_in[17];
    const int*   eidx    = (const int*)d_in[18];    // [2, E]
    const int*   batch   = (const int*)d_in[19];    // [N]
    float*       out     = (float*)d_out;           // [4096,3]

    const int* esrc = eidx;
    const int* edst = eidx + N_EDGES;

    // ---- workspace layout ----
    char* ws = (char*)d_ws;
    size_t off = 0;
    const size_t featBytes = (size_t)N_NODES * FDIM * sizeof(float);
    float* bufA = (float*)(ws + off); off += align256(featBytes);   // h
    float* bufB = (float*)(ws + off); off += align256(featBytes);   // agg / m_in
    float* bufC = (float*)(ws + off); off += align256(featBytes);   // lin1 out
    float* red  = (float*)(ws + off); off += align256(2 * FDIM * sizeof(float));
    float* vred = red + FDIM;
    float* psum = (float*)(ws + off); off += align256((size_t)N_GRAPHS * FDIM * sizeof(float));
    float* pmax = (float*)(ws + off); off += align256((size_t)N_GRAPHS * FDIM * sizeof(float));
    float* cnt  = (float*)(ws + off); off += align256((size_t)N_GRAPHS * sizeof(float));
    float* gmat = (float*)(ws + off); off += align256((size_t)N_GRAPHS * 384 * sizeof(float));

    const int mtiles    = N_NODES / 16;                  // 6250 exact
    const int gemmBlks  = (mtiles + 7) / 8;
    const int colBlks   = (N_NODES + COL_ROWS - 1) / COL_ROWS;
    const int applyBlks = (N_NODES + APPLY_ROWS - 1) / APPLY_ROWS;

    // Helper macro for one GraphNorm (+leaky) in-place on `buf`
    #define GRAPH_NORM(buf, wv, bv, sv)                                              \
        do {                                                                         \
            hipMemsetAsync(red, 0, 2 * FDIM * sizeof(float), stream);                \
            colsum_k<<<colBlks, FDIM, 0, stream>>>(buf, red, N_NODES);               \
            colvar_k<<<colBlks, FDIM, 0, stream>>>(buf, red, sv, vred, N_NODES);     \
            norm_apply_k<<<applyBlks, FDIM, 0, stream>>>(buf, red, vred,             \
                                                         wv, bv, sv, N_NODES);      \
        } while (0)

    // ======================= Layer 0 (K = 48) =======================
    {
        hipMemcpyAsync(bufB, x, (size_t)N_NODES * FIN * sizeof(float),
                       hipMemcpyDeviceToDevice, stream);
        const int groups = FIN / 4;                              // 12
        const unsigned long long tot = (unsigned long long)N_EDGES * groups;
        scatter_add_k<<<(unsigned)((tot + 255) / 256), 256, 0, stream>>>(
            x, esrc, edst, bufB, groups);

        gemm_wmma_k<<<gemmBlks, 256, 0, stream>>>(bufB, W1_0, b1_0, bufC, mtiles, FIN);
        GRAPH_NORM(bufC, gn_w + 0 * FDIM, gn_b + 0 * FDIM, gn_s + 0 * FDIM);
        gemm_wmma_k<<<gemmBlks, 256, 0, stream>>>(bufC, W2 + 0 * FDIM * FDIM,
                                                  b2 + 0 * FDIM, bufA, mtiles, FDIM);
        GRAPH_NORM(bufA, pn_w + 0 * FDIM, pn_b + 0 * FDIM, pn_s + 0 * FDIM);
    }

    // ======================= Layers 1..3 (K = 128) ==================
    for (int i = 1; i < 4; ++i) {
        hipMemcpyAsync(bufB, bufA, featBytes, hipMemcpyDeviceToDevice, stream);
        const int groups = FDIM / 4;                             // 32
        const unsigned long long tot = (unsigned long long)N_EDGES * groups;
        scatter_add_k<<<(unsigned)((tot + 255) / 256), 256, 0, stream>>>(
            bufA, esrc, edst, bufB, groups);

        gemm_wmma_k<<<gemmBlks, 256, 0, stream>>>(
            bufB, W1_rest + (size_t)(i - 1) * FDIM * FDIM,
            b1_rest + (size_t)(i - 1) * FDIM, bufC, mtiles, FDIM);
        GRAPH_NORM(bufC, gn_w + (size_t)i * FDIM, gn_b + (size_t)i * FDIM,
                   gn_s + (size_t)i * FDIM);
        gemm_wmma_k<<<gemmBlks, 256, 0, stream>>>(
            bufC, W2 + (size_t)i * FDIM * FDIM, b2 + (size_t)i * FDIM,
            bufA, mtiles, FDIM);
        if (i < 3) {
            GRAPH_NORM(bufA, pn_w + (size_t)i * FDIM, pn_b + (size_t)i * FDIM,
                       pn_s + (size_t)i * FDIM);
        }
    }

    // ======================= Pooling + readout ======================
    hipMemsetAsync(psum, 0, (size_t)N_GRAPHS * FDIM * sizeof(float), stream);
    hipMemsetAsync(cnt, 0, (size_t)N_GRAPHS * sizeof(float), stream);
    {
        const int n = N_GRAPHS * FDIM;
        fill_neg_inf_k<<<(n + 255) / 256, 256, 0, stream>>>(pmax, n);
    }
    pool_k<<<N_NODES, FDIM, 0, stream>>>(bufA, batch, psum, pmax, cnt);
    finalize_k<<<N_GRAPHS, FDIM, 0, stream>>>(psum, pmax, cnt, gmat);
    readout_k<<<N_GRAPHS, 64, 0, stream>>>(gmat, fc1_W, fc1_b, fc2_W, fc2_b, out);

    #undef GRAPH_NORM
    (void)off;
}